// GraphEncoder_70712341561346
// MI455X (gfx1250) — compile-verified
//
#include <hip/hip_runtime.h>

#define DIM 128
#define BN_EPS 1e-5f

typedef float v2f __attribute__((ext_vector_type(2)));
typedef float v8f __attribute__((ext_vector_type(8)));
typedef unsigned int u32x4 __attribute__((ext_vector_type(4)));
typedef int i32x8 __attribute__((ext_vector_type(8)));
typedef int i32x4 __attribute__((ext_vector_type(4)));

__device__ __forceinline__ void atomAddF(float* p, float v) {
    __hip_atomic_fetch_add(p, v, __ATOMIC_RELAXED, __HIP_MEMORY_SCOPE_AGENT);
}
__device__ __forceinline__ void atomAddU(unsigned* p, unsigned v) {
    __hip_atomic_fetch_add(p, v, __ATOMIC_RELAXED, __HIP_MEMORY_SCOPE_AGENT);
}

// ---------------- init: deg = 1 (self loop), BN accumulators = 0 ----------------
__global__ void k_init(unsigned* __restrict__ deg, float* __restrict__ csum,
                       float* __restrict__ csum2, int n) {
    int i = blockIdx.x * blockDim.x + threadIdx.x;
    if (i < n) deg[i] = 1u;
    if (i < DIM) { csum[i] = 0.0f; csum2[i] = 0.0f; }
}

// Build a 2-D TDM descriptor pair (D# group0 + group1), ISA cdna5 §8.3/§8.4.
// data_size=4B (code 2). Optional LDS padding (load only).
__device__ __forceinline__ void tdm_load_2d(const void* gptr, void* lptr,
                                            unsigned tensor_d0, unsigned tensor_d1,
                                            unsigned tile_d0, unsigned tile_d1,
                                            unsigned stride0,
                                            unsigned pad_en, unsigned pad_interval,
                                            unsigned pad_amount) {
    unsigned long long ga = (unsigned long long)gptr;
    unsigned lds = (unsigned)(size_t)lptr; // LDS aperture: byte offset in addr[31:0]
    u32x4 g0;
    g0[0] = 1u;                                   // count=1, is_restore=0, gather off
    g0[1] = lds;                                  // [63:32] lds_addr
    g0[2] = (unsigned)ga;                         // [95:64] global_addr[31:0]
    g0[3] = ((unsigned)(ga >> 32) & 0x01FFFFFFu)  // [120:96] global_addr[56:32]
            | (2u << 30);                         // [127:126] type=2 ("image")
    i32x8 g1;
    g1[0] = (int)((2u << 16)                      // [17:16] data_size = 4B
                  | (pad_en << 20)                // [20] pad_enable
                  | (pad_interval << 22)          // [24:22] pad_interval
                  | (pad_amount << 25));          // [31:25] pad_amount
    g1[1] = (int)((tensor_d0 & 0xFFFFu) << 16);   // [79:48] tensor_dim0 (low 16)
    g1[2] = (int)(((tensor_d0 >> 16) & 0xFFFFu)   //          tensor_dim0 (high 16)
                  | ((tensor_d1 & 0xFFFFu) << 16)); // [111:80] tensor_dim1 (low 16)
    g1[3] = (int)(((tensor_d1 >> 16) & 0xFFFFu)
                  | ((tile_d0 & 0xFFFFu) << 16)); // [127:112] tile_dim0
    g1[4] = (int)(tile_d1 & 0xFFFFu);             // [143:128] tile_dim1 ([159:144] tile_dim2=0)
    g1[5] = (int)stride0;                         // [207:160] tensor_dim0_stride (low 32)
    g1[6] = 0;                                    // stride0 hi / stride1 lo
    g1[7] = 0;
    i32x4 z4 = {0, 0, 0, 0};                      // groups 2/3 unused (2-D tensor)
    i32x8 z8 = {0, 0, 0, 0, 0, 0, 0, 0};
    __builtin_amdgcn_tensor_load_to_lds(g0, g1, z4, z4, z8, 0);
}

// ---------------- x = feature @ W via V_WMMA_F32_16X16X4_F32 ----------------
// Block = 256 threads (8 waves). Block computes 16 rows x 128 cols.
// TDM stages W (64KB) and the 16x128 A tile (LDS row stride 132 via TDM padding).
__global__ __launch_bounds__(256) void k_gemm(const float* __restrict__ F,
                                              const float* __restrict__ W,
                                              float* __restrict__ X) {
    __shared__ __align__(16) float Wl[DIM * DIM];   // 64 KB, row stride 128
    __shared__ __align__(16) float Al[16 * 132];    // 16 rows, stride 132 (conflict-free A reads)

    const int t = threadIdx.x;
    const int row0 = blockIdx.x * 16;

    // Wave 0 issues both DMAs; readfirstlane makes the guard scalar so the
    // other 7 waves branch around (TDM ignores EXEC — masking is not enough).
    int wv = __builtin_amdgcn_readfirstlane(t >> 5);
    if (wv == 0) {
        // W: full 128x128 tile, contiguous, no padding
        tdm_load_2d(W, Wl, DIM, DIM, DIM, DIM, DIM, 0u, 0u, 0u);
        // A: 16x128 tile; pad 4 DWORDs (code 3) every 128 DWORDs (code 6) -> stride 132
        tdm_load_2d(F + (size_t)row0 * DIM, Al, DIM, 16u, DIM, 16u, DIM, 1u, 6u, 3u);
        __builtin_amdgcn_s_wait_tensorcnt(0);
    }
    __syncthreads();

    const int lane = t & 31;
    const int wave = t >> 5;
    const int m    = lane & 15;                 // A: row M ; B/C: col N
    const int koff = (lane < 16) ? 0 : 2;       // K sub-offset per ISA 16x4 f32 layout
    const int n0   = wave * 16;

    v8f c = {};
#pragma unroll
    for (int k = 0; k < DIM; k += 4) {
        // A(16x4): VGPR0/1 = (M, k+koff), (M, k+koff+1)
        v2f a = *(const v2f*)&Al[m * 132 + k + koff];
        // B(4x16): VGPR0/1 = (k+koff, N), (k+koff+1, N)
        v2f b;
        b.x = Wl[(k + koff) * DIM + n0 + m];
        b.y = Wl[(k + koff + 1) * DIM + n0 + m];
        c = __builtin_amdgcn_wmma_f32_16x16x4_f32(false, a, false, b,
                                                  (short)0, c, false, false);
    }

    // C/D: VGPR r -> M = r (lanes 0-15) or r+8 (lanes 16-31), N = lane%16
    const int mbase = row0 + ((lane < 16) ? 0 : 8);
#pragma unroll
    for (int r = 0; r < 8; ++r)
        X[(long long)(mbase + r) * DIM + n0 + m] = c[r];
}

// Scalar tail for row counts not divisible by 16 (not used for N=100000, safety only)
__global__ void k_gemm_tail(const float* __restrict__ F, const float* __restrict__ W,
                            float* __restrict__ X, int rstart, int n) {
    int i = blockIdx.x * blockDim.x + threadIdx.x;
    int r = rstart + (i >> 7);
    if (r >= n) return;
    int d = i & 127;
    float acc = 0.0f;
    for (int k = 0; k < DIM; ++k) acc += F[(long long)r * DIM + k] * W[k * DIM + d];
    X[(long long)r * DIM + d] = acc;
}

// ---------------- degree over targets ----------------
__global__ void k_deg(const int* __restrict__ col, unsigned* __restrict__ deg,
                      int E, int n) {
    int e = blockIdx.x * blockDim.x + threadIdx.x;
    if (e >= E) return;
    int c = col[e];
    if ((unsigned)c < (unsigned)n) atomAddU(&deg[c], 1u);
}

__global__ void k_dinv(const unsigned* __restrict__ deg, float* __restrict__ dinv, int n) {
    int i = blockIdx.x * blockDim.x + threadIdx.x;
    if (i >= n) return;
    float d = (float)deg[i];
    dinv[i] = (d > 0.0f) ? rsqrtf(d) : 0.0f;
}

// ---------------- out = x * dinv^2 + b  (self-loop term; also initializes d_out) ----
__global__ void k_selfinit(const float* __restrict__ X, const float* __restrict__ dinv,
                           const float* __restrict__ bias, float* __restrict__ out,
                           long long total) {
    long long i = (long long)blockIdx.x * blockDim.x + threadIdx.x;
    if (i >= total) return;
    int r = (int)(i >> 7), d = (int)(i & 127);
    float di = dinv[r];
    out[i] = X[i] * (di * di) + bias[d];
}

// ---------------- edge scatter: out[col] += x[row] * dinv[row]*dinv[col] ----------
// One wave per edge: float4 gather per lane, 4 f32 atomics (resolve in L2; out fits in 192MB L2)
__global__ __launch_bounds__(256) void k_scatter(const float* __restrict__ X,
                                                 const float* __restrict__ dinv,
                                                 const int* __restrict__ rowi,
                                                 const int* __restrict__ coli,
                                                 float* __restrict__ out, int E, int n) {
    long long tid = (long long)blockIdx.x * blockDim.x + threadIdx.x;
    int e = (int)(tid >> 5);
    if (e >= E) return;
    int lane = (int)(tid & 31);
    int r = rowi[e], c = coli[e];
    if ((unsigned)r >= (unsigned)n || (unsigned)c >= (unsigned)n) return;
    float nrm = dinv[r] * dinv[c];
    const float4 v = *(const float4*)&X[(long long)r * DIM + lane * 4];
    float* dst = &out[(long long)c * DIM + lane * 4];
    atomAddF(dst + 0, v.x * nrm);
    atomAddF(dst + 1, v.y * nrm);
    atomAddF(dst + 2, v.z * nrm);
    atomAddF(dst + 3, v.w * nrm);
}

// ---------------- per-column sums for BN (register accumulate, 2 atomics/thread) ----
#define ROWS_PER_BLOCK 100
__global__ __launch_bounds__(256) void k_bnstats(const float* __restrict__ out,
                                                 float* __restrict__ csum,
                                                 float* __restrict__ csum2, int n) {
    int c = threadIdx.x & 127;
    int half = threadIdx.x >> 7; // 0 or 1
    int r0 = blockIdx.x * ROWS_PER_BLOCK + half;
    int rend = min(blockIdx.x * ROWS_PER_BLOCK + ROWS_PER_BLOCK, n);
    float s = 0.0f, s2 = 0.0f;
    for (int r = r0; r < rend; r += 2) {
        int rp = min(r + 8, n - 1);
        __builtin_prefetch(&out[(long long)rp * DIM + c], 0, 0);
        float v = out[(long long)r * DIM + c];
        s += v;
        s2 += v * v;
    }
    atomAddF(&csum[c], s);
    atomAddF(&csum2[c], s2);
}

// ---------------- BN apply (in place on d_out) ----------------
__global__ void k_bnapply(float* __restrict__ out, const float* __restrict__ csum,
                          const float* __restrict__ csum2, const float* __restrict__ bnw,
                          const float* __restrict__ bnb, long long total, float invN) {
    long long i = (long long)blockIdx.x * blockDim.x + threadIdx.x;
    if (i >= total) return;
    int c = (int)(i & 127);
    float mean = csum[c] * invN;
    float var = csum2[c] * invN - mean * mean;
    float istd = rsqrtf(var + BN_EPS);
    out[i] = (out[i] - mean) * istd * bnw[c] + bnb[c];
}

static inline size_t alignUp(size_t x) { return (x + 255) & ~(size_t)255; }

extern "C" void kernel_launch(void* const* d_in, const int* in_sizes, int n_in,
                              void* d_out, int out_size, void* d_ws, size_t ws_size,
                              hipStream_t stream) {
    const float* feature = (const float*)d_in[0];
    const int*   ei      = (const int*)d_in[1]; // [2, E] int32
    const float* W       = (const float*)d_in[2];
    const float* bias    = (const float*)d_in[3];
    const float* bnw     = (const float*)d_in[4];
    const float* bnb     = (const float*)d_in[5];
    float* out = (float*)d_out;

    const int N = in_sizes[0] / DIM;
    const int E = in_sizes[1] / 2;
    const long long total = (long long)N * DIM;

    // workspace layout
    char* ws = (char*)d_ws;
    size_t o = 0;
    float*    x    = (float*)(ws + o);    o = alignUp(o + (size_t)total * 4);
    unsigned* deg  = (unsigned*)(ws + o); o = alignUp(o + (size_t)N * 4);
    float*    dinv = (float*)(ws + o);    o = alignUp(o + (size_t)N * 4);
    float*    csum = (float*)(ws + o);    o = alignUp(o + DIM * 4);
    float*    csum2= (float*)(ws + o);    o = alignUp(o + DIM * 4);
    (void)ws_size; (void)n_in; (void)out_size;

    const int B = 256;

    // 1) init deg / BN accumulators
    k_init<<<(N + B - 1) / B, B, 0, stream>>>(deg, csum, csum2, N);

    // 2) x = feature @ W (WMMA fp32, TDM-staged tiles)
    int nTiles = N / 16;
    if (nTiles > 0)
        k_gemm<<<nTiles, B, 0, stream>>>(feature, W, x);
    int rem = N - nTiles * 16;
    if (rem > 0) {
        long long tailThreads = (long long)rem * DIM;
        k_gemm_tail<<<(int)((tailThreads + B - 1) / B), B, 0, stream>>>(
            feature, W, x, nTiles * 16, N);
    }

    // 3) degree over targets, 4) dinv_sqrt
    k_deg<<<(E + B - 1) / B, B, 0, stream>>>(ei + E, deg, E, N);
    k_dinv<<<(N + B - 1) / B, B, 0, stream>>>(deg, dinv, N);

    // 5) self-loop term + bias (initializes out)
    k_selfinit<<<(int)((total + B - 1) / B), B, 0, stream>>>(x, dinv, bias, out, total);

    // 6) edge scatter (one wave per edge)
    long long scatThreads = (long long)E * 32;
    k_scatter<<<(int)((scatThreads + B - 1) / B), B, 0, stream>>>(
        x, dinv, ei, ei + E, out, E, N);

    // 7) BN stats, 8) BN apply
    int statBlocks = (N + ROWS_PER_BLOCK - 1) / ROWS_PER_BLOCK;
    k_bnstats<<<statBlocks, B, 0, stream>>>(out, csum, csum2, N);
    k_bnapply<<<(int)((total + B - 1) / B), B, 0, stream>>>(
        out, csum, csum2, bnw, bnb, total, 1.0f / (float)N);
}